// DGLGCNModel_11081015623722
// MI455X (gfx1250) — compile-verified
//
#include <hip/hip_runtime.h>
#include <stdint.h>

#define N_NODES 50000
#define FEAT    128

typedef float v2f __attribute__((ext_vector_type(2)));
typedef float v8f __attribute__((ext_vector_type(8)));

// ---------------- utility kernels ----------------

__global__ void zero_f32(float* __restrict__ p, int n) {
    int i = blockIdx.x * blockDim.x + threadIdx.x;
    if (i < n) p[i] = 0.0f;
}

__global__ void deg_kernel(const long long* __restrict__ dst, int E,
                           float* __restrict__ deg) {
    int e = blockIdx.x * blockDim.x + threadIdx.x;
    if (e < E) atomicAdd(&deg[dst[e]], 1.0f);
}

__global__ void invdeg_kernel(const float* __restrict__ deg,
                              float* __restrict__ inv, int n) {
    int i = blockIdx.x * blockDim.x + threadIdx.x;
    if (i < n) inv[i] = 1.0f / fmaxf(deg[i], 1.0f);
}

// One wave32 per edge; 8 edges per 256-thread block.
// Lane l handles features {l, l+32, l+64, l+96}: every global_load_b32 and
// every global_atomic_add_f32 instruction covers 32 consecutive floats
// (128 B contiguous) so the L2 atomic units get dense 64B-granule requests.
__global__ void scatter_add_edges(const float* __restrict__ x,
                                  const long long* __restrict__ src,
                                  const long long* __restrict__ dst,
                                  float* __restrict__ agg, int E) {
    const int lane = threadIdx.x & 31;
    const int wv   = threadIdx.x >> 5;
    const int e    = blockIdx.x * 8 + wv;
    if (e >= E) return;

    const int s = (int)src[e];
    const int d = (int)dst[e];
    const float* __restrict__ xr = x   + (size_t)s * FEAT;
    float*       __restrict__ ar = agg + (size_t)d * FEAT;

#pragma unroll
    for (int j = 0; j < 4; ++j) {
        const int f = lane + 32 * j;
        atomicAdd(&ar[f], xr[f]);
    }
}

// ---------------- fp32 WMMA GEMM: out[N,OUT] = (agg*inv_deg) @ W + bias ----------------
// One wave32 computes one 16x16 output tile using V_WMMA_F32_16X16X4_F32.
// blockIdx.x = row tile (16 rows), wave-in-block = column tile.
//
// A fragment (16x4 f32, 2 VGPRs/lane):
//   lanes 0-15 : M=lane,    v0 = A[M][k0+0], v1 = A[M][k0+1]
//   lanes 16-31: M=lane-16, v0 = A[M][k0+2], v1 = A[M][k0+3]
// B fragment (4x16 f32) mirrors that with N=lane%16.
// C/D (16x16 f32, 8 VGPRs): row = v + (lane/16)*8, col = lane%16.
template <int OUT, bool RELU>
__global__ void gcn_gemm_wmma(const float* __restrict__ agg,
                              const float* __restrict__ invdeg,
                              const float* __restrict__ W,   // [FEAT, OUT] row-major
                              const float* __restrict__ bias,
                              float* __restrict__ out) {
    const int lane = threadIdx.x & 31;
    const int wave = threadIdx.x >> 5;

    const int row0 = blockIdx.x * 16;   // N_NODES % 16 == 0 -> no guards, EXEC all-ones
    const int col0 = wave * 16;

    const int m    = lane & 15;         // A row within tile / B-C-D column within tile
    const int kgrp = lane >> 4;         // 0: K{0,1}, 1: K{2,3}

    const int   arow  = row0 + m;
    const float scale = invdeg[arow];   // fold right-normalization into A load
    const float* __restrict__ aptr = agg + (size_t)arow * FEAT;

    v8f c = {};
#pragma unroll
    for (int k0 = 0; k0 < FEAT; k0 += 4) {
        const int ka = k0 + kgrp * 2;
        v2f a, b;
        a.x = aptr[ka]     * scale;
        a.y = aptr[ka + 1] * scale;
        b.x = W[(size_t)ka       * OUT + col0 + m];
        b.y = W[(size_t)(ka + 1) * OUT + col0 + m];
        // (neg_a, A, neg_b, B, c_mod, C, reuse_a, reuse_b)
        c = __builtin_amdgcn_wmma_f32_16x16x4_f32(false, a, false, b,
                                                  (short)0, c, false, false);
    }

    const int   ocol = col0 + m;
    const float bv   = bias[ocol];
#pragma unroll
    for (int v = 0; v < 8; ++v) {
        const int r = row0 + v + kgrp * 8;
        float val = c[v] + bv;
        if (RELU) val = fmaxf(val, 0.0f);
        out[(size_t)r * OUT + ocol] = val;
    }
}

// ---------------- launch ----------------

extern "C" void kernel_launch(void* const* d_in, const int* in_sizes, int n_in,
                              void* d_out, int out_size, void* d_ws, size_t ws_size,
                              hipStream_t stream) {
    const float*     x   = (const float*)d_in[0];
    const long long* src = (const long long*)d_in[1];
    const long long* dst = (const long long*)d_in[2];
    const float*     W1  = (const float*)d_in[3];
    const float*     b1  = (const float*)d_in[4];
    const float*     W2  = (const float*)d_in[5];
    const float*     b2  = (const float*)d_in[6];
    float*           out = (float*)d_out;

    const int E = in_sizes[1];

    // workspace layout
    float* deg  = (float*)d_ws;                       // N
    float* inv  = deg  + N_NODES;                     // N
    float* bufA = inv  + N_NODES;                     // N*128 (agg, reused per layer)
    float* bufB = bufA + (size_t)N_NODES * FEAT;      // N*128 (hidden h)

    const int NF = N_NODES * FEAT;                    // 6.4M, fits int
    const int edgeBlocks = (E + 7) / 8;               // 8 edges (waves) per block

    // degrees -> inv_deg
    zero_f32     <<<(N_NODES + 255) / 256, 256, 0, stream>>>(deg, N_NODES);
    deg_kernel   <<<(E + 255) / 256,       256, 0, stream>>>(dst, E, deg);
    invdeg_kernel<<<(N_NODES + 255) / 256, 256, 0, stream>>>(deg, inv, N_NODES);

    // layer 1: agg = scatter_add(x[src] -> dst); h = relu((agg*inv) @ W1 + b1)
    zero_f32<<<(NF + 255) / 256, 256, 0, stream>>>(bufA, NF);
    scatter_add_edges<<<edgeBlocks, 256, 0, stream>>>(x, src, dst, bufA, E);
    gcn_gemm_wmma<128, true><<<N_NODES / 16, 256, 0, stream>>>(bufA, inv, W1, b1, bufB);

    // layer 2: agg2 = scatter_add(h[src] -> dst); out = (agg2*inv) @ W2 + b2
    zero_f32<<<(NF + 255) / 256, 256, 0, stream>>>(bufA, NF);
    scatter_add_edges<<<edgeBlocks, 256, 0, stream>>>(bufB, src, dst, bufA, E);
    gcn_gemm_wmma<64, false><<<N_NODES / 16, 128, 0, stream>>>(bufA, inv, W2, b2, out);
}